// WordByWord_73615739453585
// MI455X (gfx1250) — compile-verified
//
#include <hip/hip_runtime.h>
#include <hip/hip_bf16.h>

// ---------------------------------------------------------------------------
// WordByWord matcher on MI455X (gfx1250), FP32 WMMA 16x16x4 path.
// Fused: gather -> Linear(300,300) -> LeakyReLU -> row-normalize -> cos-sim
//        -> max over text positions, all per 16-row text tile in LDS.
// Edge tiles read a zeroed workspace row instead of per-step cndmask.
// ---------------------------------------------------------------------------

typedef __attribute__((ext_vector_type(2))) float v2f;
typedef __attribute__((ext_vector_type(8))) float v8f;

#define DDIM   300
#define DPITCH 304          // LDS/row pitch (pad to 16)
#define BATCH  32
#define LTXT   2048
#define LQRY   300
#define LANS   64
#define NDT    19           // col tiles of 300 (pad to 304)
#define NQT    19           // q row tiles
#define NAT    4            // a row tiles
#define NTT    128          // text row tiles (2048/16)
#define EPS_WM 1e-7f
#define SLOPE  0.01f
#define NEGINF (-3.4e38f)

__device__ __forceinline__ float leaky(float x) { return x >= 0.f ? x : SLOPE * x; }

// One 16x16x4 f32 WMMA step; a = A[m][k0+2h .. +1], b = B[k0+2h .. +1][n].
__device__ __forceinline__ v8f wmma_f32_step(v2f a, v2f b, v8f c) {
    return __builtin_amdgcn_wmma_f32_16x16x4_f32(false, a, false, b, (short)0, c,
                                                 false, false);
}

// 8 threads per row, shuffle-tree reduce sum of squares -> sinv[16].
__device__ __forceinline__ void row_norms(const float* Es, float* sinv, int tid) {
    int r = tid >> 3, c0 = tid & 7;
    float s = 0.f;
    for (int c = c0; c < DDIM; c += 8) { float v = Es[r * DPITCH + c]; s += v * v; }
    s += __shfl_xor(s, 1, 32);
    s += __shfl_xor(s, 2, 32);
    s += __shfl_xor(s, 4, 32);
    if (c0 == 0) sinv[r] = 1.f / fmaxf(sqrtf(s), 1e-8f);
}

// ---------------------------------------------------------------------------
__global__ __launch_bounds__(320)
void zero_kernel(float* __restrict__ p, int n) {
    int i = threadIdx.x;
    if (i < n) p[i] = 0.f;
}

// ---------------------------------------------------------------------------
// Kernel A: gather tokens, project through W (out = leaky(X @ W^T + bias)),
// row-normalize, store normalized rows. Optionally emit sliding partial max
// (max over the tile's rows of slidW[row] * raw_leaky_value, per column).
// grid.x = BATCH * nTiles, block = 128 (4 waves).
// ---------------------------------------------------------------------------
__global__ __launch_bounds__(128)
void proj_norm_kernel(const int* __restrict__ tokens, const float* __restrict__ emb,
                      const float* __restrict__ W, const float* __restrict__ bias,
                      const float* __restrict__ zrow,  // 304 zeros (for edge cols)
                      int L, int nTiles, int rowPad,
                      float* __restrict__ outN,          // [B][rowPad][DDIM], padded rows = 0
                      const float* __restrict__ slidW,   // [L] or null
                      float* __restrict__ partialS)      // [B][nTiles][DPITCH] or null
{
    __shared__ float Xs[16 * DPITCH];
    __shared__ float Es[16 * DPITCH];
    __shared__ int   stok[16];
    __shared__ float sinv[16];

    const int tid  = threadIdx.x;
    const int b    = blockIdx.x / nTiles;
    const int tile = blockIdx.x % nTiles;

    if (tid < 16) {
        int row = tile * 16 + tid;
        stok[tid] = (row < L) ? tokens[b * L + row] : -1;
    }
    __syncthreads();
    for (int i = tid; i < 16 * DDIM; i += 128) {
        int r = i / DDIM, c = i - r * DDIM;
        int tk = stok[r];
        Xs[r * DPITCH + c] = (tk >= 0) ? emb[(long)tk * DDIM + c] : 0.f;
    }
    __syncthreads();

    const int wave = tid >> 5, lane = tid & 31;
    const int h = lane >> 4, nn = lane & 15;

    for (int ct = wave; ct < NDT; ct += 4) {          // uniform per wave
        int col = ct * 16 + nn;
        bool cv = col < DDIM;
        const float* wrow = cv ? (W + (long)col * DDIM) : zrow;  // no per-step mask
        v8f acc = {};
        for (int k0 = 0; k0 < DDIM; k0 += 4) {
            v2f a  = *(const v2f*)&Xs[nn * DPITCH + k0 + 2 * h];
            v2f bv = *(const v2f*)(wrow + k0 + 2 * h);
            acc = wmma_f32_step(a, bv, acc);
        }
        float bcol = cv ? bias[col] : 0.f;
#pragma unroll
        for (int j = 0; j < 8; ++j) {
            int m = j + 8 * h;
            float v = leaky(acc[j] + bcol);
            Es[m * DPITCH + col] = cv ? v : 0.f;
        }
    }
    __syncthreads();

    row_norms(Es, sinv, tid);
    __syncthreads();

    for (int i = tid; i < 16 * DDIM; i += 128) {
        int r = i / DDIM, c = i - r * DDIM;
        int row = tile * 16 + r;
        if (row < rowPad) {
            float v = (row < L) ? Es[r * DPITCH + c] * sinv[r] : 0.f;
            outN[((long)b * rowPad + row) * DDIM + c] = v;
        }
    }
    if (partialS) {
        for (int c = tid; c < DDIM; c += 128) {
            float m = NEGINF;
            for (int r = 0; r < 16; ++r) {
                int row = tile * 16 + r;
                if (row < L) m = fmaxf(m, slidW[row] * Es[r * DPITCH + c]);
            }
            partialS[((long)b * nTiles + tile) * DPITCH + c] = m;
        }
    }
}

// ---------------------------------------------------------------------------
// Kernel B (fused, dominant): per (b, text 16-row tile): gather, project via
// Wt, leaky, normalize in LDS, then WMMA cos-sim vs qN (19 tiles) and aN
// (4 tiles), emitting per-tile column maxes. grid.x = BATCH*NTT, block = 128.
// ---------------------------------------------------------------------------
__global__ __launch_bounds__(128)
void text_sim_kernel(const int* __restrict__ text, const float* __restrict__ emb,
                     const float* __restrict__ Wt, const float* __restrict__ bt,
                     const float* __restrict__ zrow,
                     const float* __restrict__ qN,   // [B][DPITCH][DDIM]
                     const float* __restrict__ aN,   // [B][LANS][DDIM]
                     float* __restrict__ partialQ,   // [B][NTT][DPITCH]
                     float* __restrict__ partialA)   // [B][NTT][LANS]
{
    __shared__ float Xs[16 * DPITCH];
    __shared__ float Es[16 * DPITCH];
    __shared__ int   stok[16];
    __shared__ float sinv[16];

    const int tid  = threadIdx.x;
    const int b    = blockIdx.x / NTT;
    const int tile = blockIdx.x % NTT;

    if (tid < 16) stok[tid] = text[b * LTXT + tile * 16 + tid];
    __syncthreads();
    for (int i = tid; i < 16 * DDIM; i += 128) {
        int r = i / DDIM, c = i - r * DDIM;
        Xs[r * DPITCH + c] = emb[(long)stok[r] * DDIM + c];
    }
    __syncthreads();

    const int wave = tid >> 5, lane = tid & 31;
    const int h = lane >> 4, nn = lane & 15;

    // ---- projection: Es = leaky(Xs @ Wt^T + bt) ----
    for (int ct = wave; ct < NDT; ct += 4) {
        int col = ct * 16 + nn;
        bool cv = col < DDIM;
        const float* wrow = cv ? (Wt + (long)col * DDIM) : zrow;
        v8f acc = {};
        for (int k0 = 0; k0 < DDIM; k0 += 4) {
            v2f a  = *(const v2f*)&Xs[nn * DPITCH + k0 + 2 * h];
            v2f bv = *(const v2f*)(wrow + k0 + 2 * h);
            acc = wmma_f32_step(a, bv, acc);
        }
        float bcol = cv ? bt[col] : 0.f;
#pragma unroll
        for (int j = 0; j < 8; ++j) {
            int m = j + 8 * h;
            float v = leaky(acc[j] + bcol);
            Es[m * DPITCH + col] = cv ? v : 0.f;
        }
    }
    __syncthreads();

    // ---- row norms + in-place normalize ----
    row_norms(Es, sinv, tid);
    __syncthreads();
    for (int i = tid; i < 16 * DPITCH; i += 128) Es[i] *= sinv[i / DPITCH];
    __syncthreads();

    // ---- cos-sim tiles vs qN (0..18) and aN (19..22), max over 16 t-rows ----
    for (int t = wave; t < NQT + NAT; t += 4) {       // uniform per wave
        bool isQ = t < NQT;
        int tn   = isQ ? t : t - NQT;
        int col  = tn * 16 + nn;                      // q (0..303) or a (0..63) index
        const float* nrow = isQ ? (qN + ((long)b * DPITCH + col) * DDIM)
                                : (aN + ((long)b * LANS  + col) * DDIM);
        v8f acc = {};
        for (int k0 = 0; k0 < DDIM; k0 += 4) {
            v2f a  = *(const v2f*)&Es[nn * DPITCH + k0 + 2 * h];
            v2f bv = *(const v2f*)(nrow + k0 + 2 * h);
            acc = wmma_f32_step(a, bv, acc);
        }
        float mx = NEGINF;
#pragma unroll
        for (int j = 0; j < 8; ++j) mx = fmaxf(mx, acc[j]);   // over M = j+8h
        mx = fmaxf(mx, __shfl_xor(mx, 16, 32));               // combine halves
        if (h == 0) {
            if (isQ) partialQ[((long)b * NTT + tile) * DPITCH + col] = mx;
            else     partialA[((long)b * NTT + tile) * LANS   + col] = mx;
        }
    }
}

// ---------------------------------------------------------------------------
// Kernel C: reduce partial maxes over tiles. grid.x = BATCH, block = 256.
// ---------------------------------------------------------------------------
__global__ __launch_bounds__(256)
void reduce_kernel(const float* __restrict__ partialQ, const float* __restrict__ partialA,
                   const float* __restrict__ partialS,
                   float* __restrict__ maxQ, float* __restrict__ maxA,
                   float* __restrict__ maxQs)
{
    const int b = blockIdx.x, tid = threadIdx.x;
    for (int q = tid; q < DDIM; q += 256) {
        float m = NEGINF;
        for (int t = 0; t < NTT; ++t) m = fmaxf(m, partialQ[((long)b * NTT + t) * DPITCH + q]);
        maxQ[b * DDIM + q] = m;
    }
    for (int a = tid; a < LANS; a += 256) {
        float m = NEGINF;
        for (int t = 0; t < NTT; ++t) m = fmaxf(m, partialA[((long)b * NTT + t) * LANS + a]);
        maxA[b * LANS + a] = m;
    }
    for (int d = tid; d < DDIM; d += 256) {
        float m = NEGINF;
        for (int t = 0; t < NQT; ++t) m = fmaxf(m, partialS[((long)b * NQT + t) * DPITCH + d]);
        maxQs[b * DDIM + d] = m;
    }
}

// ---------------------------------------------------------------------------
// Kernel D: weighted means + alpha combine. 1 block, 32 threads (one per b).
// ---------------------------------------------------------------------------
__global__ __launch_bounds__(32)
void final_kernel(const float* __restrict__ maxQ, const float* __restrict__ maxA,
                  const float* __restrict__ maxQs,
                  const float* __restrict__ qW, const float* __restrict__ aW,
                  const float* __restrict__ alphaQ, const float* __restrict__ alphaA,
                  const float* __restrict__ alphaQA, float* __restrict__ out)
{
    const int b = threadIdx.x;
    if (b >= BATCH) return;
    float sq = 0.f, sa = 0.f;
    for (int i = 0; i < LQRY; ++i) sq += qW[i];
    for (int i = 0; i < LANS; ++i) sa += aW[i];
    float dq = 0.f, da = 0.f, ds = 0.f;
    for (int i = 0; i < LQRY; ++i) dq += maxQ[b * DDIM + i] * qW[i];
    for (int i = 0; i < LANS; ++i) da += maxA[b * LANS + i] * aW[i];
    for (int i = 0; i < DDIM; ++i) ds += maxQs[b * DDIM + i] * qW[i];
    float Mq = dq / (sq + EPS_WM);
    float Ma = da / (sa + EPS_WM);
    float Ms = ds / (sq + EPS_WM);
    float aQ = alphaQ[0], aA = alphaA[0], aQA = alphaQA[0];
    out[b]         = aQ * Mq + aA * Ma + aQA * Mq * Ma;   // Mword
    out[BATCH + b] = aQ * Ms + aA * Ma + aQA * Ms * Ma;   // Mssw
}

// ---------------------------------------------------------------------------
extern "C" void kernel_launch(void* const* d_in, const int* in_sizes, int n_in,
                              void* d_out, int out_size, void* d_ws, size_t ws_size,
                              hipStream_t stream) {
    (void)in_sizes; (void)n_in; (void)out_size; (void)ws_size;
    const int*   question = (const int*)  d_in[0];
    const int*   text     = (const int*)  d_in[1];
    const int*   answer   = (const int*)  d_in[2];
    const float* emb      = (const float*)d_in[3];
    const float* Wt       = (const float*)d_in[4];
    const float* bt       = (const float*)d_in[5];
    const float* Wq       = (const float*)d_in[6];
    const float* bq       = (const float*)d_in[7];
    const float* Wa       = (const float*)d_in[8];
    const float* ba       = (const float*)d_in[9];
    const float* qW       = (const float*)d_in[10];
    const float* aW       = (const float*)d_in[11];
    const float* slidW    = (const float*)d_in[12];
    const float* alphaQ   = (const float*)d_in[13];
    const float* alphaA   = (const float*)d_in[14];
    const float* alphaQA  = (const float*)d_in[15];

    float* ws = (float*)d_ws;
    size_t o = 0;
    float* qN       = ws + o; o += (size_t)BATCH * DPITCH * DDIM;   // 2,918,400
    float* aN       = ws + o; o += (size_t)BATCH * LANS   * DDIM;   //   614,400
    float* partialS = ws + o; o += (size_t)BATCH * NQT * DPITCH;    //   184,832
    float* partialQ = ws + o; o += (size_t)BATCH * NTT * DPITCH;    // 1,245,184
    float* partialA = ws + o; o += (size_t)BATCH * NTT * LANS;      //   262,144
    float* maxQ     = ws + o; o += (size_t)BATCH * DDIM;
    float* maxA     = ws + o; o += (size_t)BATCH * LANS;
    float* maxQs    = ws + o; o += (size_t)BATCH * DDIM;
    float* zrow     = ws + o; o += (size_t)DPITCH;                  // 304 zeros

    zero_kernel<<<1, 320, 0, stream>>>(zrow, DPITCH);
    proj_norm_kernel<<<BATCH * NQT, 128, 0, stream>>>(
        question, emb, Wq, bq, zrow, LQRY, NQT, DPITCH, qN, slidW, partialS);
    proj_norm_kernel<<<BATCH * NAT, 128, 0, stream>>>(
        answer, emb, Wa, ba, zrow, LANS, NAT, LANS, aN, nullptr, nullptr);
    text_sim_kernel<<<BATCH * NTT, 128, 0, stream>>>(
        text, emb, Wt, bt, zrow, qN, aN, partialQ, partialA);
    reduce_kernel<<<BATCH, 256, 0, stream>>>(
        partialQ, partialA, partialS, maxQ, maxA, maxQs);
    final_kernel<<<1, 32, 0, stream>>>(
        maxQ, maxA, maxQs, qW, aW, alphaQ, alphaA, alphaQA, (float*)d_out);
}